// LearnerC_Aggregator_12120397709897
// MI455X (gfx1250) — compile-verified
//
#include <hip/hip_runtime.h>
#include <hip/hip_bf16.h>

typedef float v2f __attribute__((ext_vector_type(2)));
typedef float v8f __attribute__((ext_vector_type(8)));

#define EMBED_DIM       32
#define KDIM            96      // 3 * EMBED_DIM
#define XS_STRIDE       100     // padded LDS row stride (bank-conflict free)
#define WS_STRIDE       100
#define NODES_PER_BLOCK 128
#define THREADS         256

__global__ __launch_bounds__(THREADS)
void learnerc_aggregator_wmma(const int* __restrict__ nodes,
                              const int* __restrict__ gender,
                              const int* __restrict__ age,
                              const float* __restrict__ u_emb,
                              const float* __restrict__ g_emb,
                              const float* __restrict__ a_emb,
                              const float* __restrict__ W,      // [32, 96] row-major
                              const float* __restrict__ bias,   // [32]
                              float* __restrict__ out,          // [N, 32]
                              int n_nodes)
{
    __shared__ float Xs[NODES_PER_BLOCK * XS_STRIDE]; // 128 x 96 feature tile (padded)
    __shared__ float Ws[32 * WS_STRIDE];              // 32 x 96 weight tile (padded)

    const int tid        = threadIdx.x;
    const int block_base = blockIdx.x * NODES_PER_BLOCK;

    // ---- Stage W (32x96) into LDS, float4 vectorized -----------------------
    for (int q = tid; q < (32 * KDIM) / 4; q += THREADS) {
        const int flat = q * 4;
        const int row  = flat / KDIM;
        const int col  = flat % KDIM;
        const float4 wv = *(const float4*)(W + row * KDIM + col);
        *(float4*)(&Ws[row * WS_STRIDE + col]) = wv;
    }

    // ---- Gather phase: 3 segments x 128 nodes = 384 row-copy tasks ---------
    for (int task = tid; task < 3 * NODES_PER_BLOCK; task += THREADS) {
        const int seg   = task >> 7;       // 0: gender-emb, 1: age-emb, 2: user-emb
        const int nl    = task & 127;
        const int gnode = block_base + nl;
        const float* src;
        if (gnode < n_nodes) {
            const int uid = nodes[gnode];
            if (seg == 0)      src = g_emb + (long)gender[uid] * EMBED_DIM;
            else if (seg == 1) src = a_emb + (long)age[uid]    * EMBED_DIM;
            else               src = u_emb + (long)uid         * EMBED_DIM;
        } else {
            src = g_emb; // dummy row; result discarded by store guard
        }
        float* dst = &Xs[nl * XS_STRIDE + seg * EMBED_DIM];
        #pragma unroll
        for (int v = 0; v < EMBED_DIM / 4; ++v)
            ((float4*)dst)[v] = ((const float4*)src)[v];
    }
    __syncthreads();

    // ---- WMMA GEMM: per-wave 16-node tile, K = 96 in 24 steps of 4 ---------
    const int wave = tid >> 5;          // 8 waves (wave32)
    const int lane = tid & 31;
    const int m0   = wave * 16;         // node sub-tile within block
    const int mrow = lane & 15;
    const int hi   = lane >> 4;         // half-wave selector (K split)

    v8f acc0 = {0.f, 0.f, 0.f, 0.f, 0.f, 0.f, 0.f, 0.f}; // out dims 0..15
    v8f acc1 = {0.f, 0.f, 0.f, 0.f, 0.f, 0.f, 0.f, 0.f}; // out dims 16..31

    const float* xrow  = &Xs[(m0 + mrow) * XS_STRIDE];   // A: M = lane&15
    const float* wrow0 = &Ws[(mrow)      * WS_STRIDE];   // B: N = lane&15, out-tile 0
    const float* wrow1 = &Ws[(16 + mrow) * WS_STRIDE];   // B: out-tile 1

    #pragma unroll
    for (int kk = 0; kk < KDIM / 4; ++kk) {
        const int kc = kk * 4 + 2 * hi;  // K pair held by this half-wave
        const v2f a  = *(const v2f*)(xrow  + kc);
        const v2f b0 = *(const v2f*)(wrow0 + kc);
        const v2f b1 = *(const v2f*)(wrow1 + kc);
        acc0 = __builtin_amdgcn_wmma_f32_16x16x4_f32(false, a, false, b0,
                                                     (short)0, acc0, false, false);
        acc1 = __builtin_amdgcn_wmma_f32_16x16x4_f32(false, a, false, b1,
                                                     (short)0, acc1, false, false);
    }

    // ---- Epilogue: bias + ReLU, store D ------------------------------------
    // D layout: VGPR r, lane l -> [M = r + 8*(l>>4)][N = l&15]
    const int nd0        = lane & 15;
    const float bv0      = bias[nd0];
    const float bv1      = bias[16 + nd0];
    const int gnode_base = block_base + m0 + 8 * hi;

    #pragma unroll
    for (int r = 0; r < 8; ++r) {
        const int gnode = gnode_base + r;
        if (gnode < n_nodes) {
            float v0 = acc0[r] + bv0; v0 = v0 > 0.f ? v0 : 0.f;
            float v1 = acc1[r] + bv1; v1 = v1 > 0.f ? v1 : 0.f;
            out[(long)gnode * EMBED_DIM + nd0]      = v0;
            out[(long)gnode * EMBED_DIM + 16 + nd0] = v1;
        }
    }
}

extern "C" void kernel_launch(void* const* d_in, const int* in_sizes, int n_in,
                              void* d_out, int out_size, void* d_ws, size_t ws_size,
                              hipStream_t stream) {
    const int*   nodes  = (const int*)d_in[0];
    const int*   gender = (const int*)d_in[1];
    const int*   age    = (const int*)d_in[2];
    const float* u_emb  = (const float*)d_in[3];
    const float* g_emb  = (const float*)d_in[4];
    const float* a_emb  = (const float*)d_in[5];
    const float* W      = (const float*)d_in[6];
    const float* bias   = (const float*)d_in[7];
    float*       out    = (float*)d_out;

    const int n_nodes = in_sizes[0];
    const int blocks  = (n_nodes + NODES_PER_BLOCK - 1) / NODES_PER_BLOCK;

    learnerc_aggregator_wmma<<<blocks, THREADS, 0, stream>>>(
        nodes, gender, age, u_emb, g_emb, a_emb, W, bias, out, n_nodes);
}